// GraphRecommendationModel_68410239091398
// MI455X (gfx1250) — compile-verified
//
#include <hip/hip_runtime.h>
#include <hip/hip_bf16.h>

// ---------------- problem constants (match reference) ----------------
#define NU_ 100000   // users
#define NB_ 200000   // books
#define NN_ 300000   // total nodes
#define D_  128      // embedding dim
#define F_  512      // book feature dim
#define E_  600000   // graph edges
#define P_  200000   // prediction edges
#define H_  64       // predictor hidden

typedef __attribute__((ext_vector_type(2))) float v2f;
typedef __attribute__((ext_vector_type(8))) float v8f;

// ---------------- small utility kernels ----------------
__global__ __launch_bounds__(256) void k_zero(float* __restrict__ p, long n) {
    long i = (long)blockIdx.x * blockDim.x + threadIdx.x;
    if (i < n) p[i] = 0.0f;
}

__global__ __launch_bounds__(256) void k_deg(const int* __restrict__ dst,
                                             float* __restrict__ deg, int e) {
    int i = blockIdx.x * blockDim.x + threadIdx.x;
    if (i < e) atomicAdd(&deg[dst[i]], 1.0f);
}

__global__ __launch_bounds__(256) void k_dinv(float* __restrict__ deg, int n) {
    int i = blockIdx.x * blockDim.x + threadIdx.x;
    if (i < n) deg[i] = rsqrtf(deg[i] + 1.0f);   // +1 = self loop; always > 0
}

// ---------------- WMMA f32 GEMM: C[M,128] = A[M,K] @ B[K,128] (+bias) ----------------
// One wave computes one 16x16 tile; 8 waves per block cover 16 rows x 128 cols,
// sharing the A row-tile through LDS (stride K+4 -> conflict-free column reads).
template <int K, bool RELU, bool HASBIAS>
__global__ __launch_bounds__(256) void gemm16_wmma(const float* __restrict__ A,
                                                   const float* __restrict__ B,
                                                   const float* __restrict__ bias,
                                                   float* __restrict__ C) {
    constexpr int LDK = K + 4;                 // pad keeps 16B align, kills bank conflicts
    __shared__ float sA[16 * LDK];

    const int  tid  = threadIdx.x;
    const long row0 = (long)blockIdx.x * 16;

    // cooperative load of the 16 x K A-tile (float4 granularity)
    constexpr int NV4 = 16 * (K / 4);
    for (int i = tid; i < NV4; i += 256) {
        const int r  = i / (K / 4);
        const int c4 = i % (K / 4);
        float4 v = *(const float4*)(A + (row0 + r) * (long)K + (long)c4 * 4);
        *(float4*)(&sA[r * LDK + c4 * 4]) = v;
    }
    __syncthreads();

    const int lane = tid & 31;
    const int wave = tid >> 5;      // 0..7 -> column tile
    const int hi   = lane >> 4;     // half-wave selects K pair (and C row + 8)
    const int lo   = lane & 15;     // A row / B,C column within tile
    const int n0   = wave << 4;

    // A frag: element (m=lo, k = kbase + 2*hi + j) in vgpr j
    const float* sAp = &sA[lo * LDK + 2 * hi];
    // B frag: element (k = kbase + 2*hi + j, n = n0+lo) in vgpr j
    const float* Bp = B + (long)(2 * hi) * D_ + n0 + lo;

    v8f c = {};
#pragma unroll 4
    for (int k = 0; k < K; k += 4) {
        v2f a = *(const v2f*)(sAp + k);        // ds_load_b64
        v2f b;
        b[0] = Bp[(long)k * D_];
        b[1] = Bp[(long)k * D_ + D_];
        c = __builtin_amdgcn_wmma_f32_16x16x4_f32(false, a, false, b,
                                                  (short)0, c, false, false);
    }

    const float bv = HASBIAS ? bias[n0 + lo] : 0.0f;
    // C/D layout: vgpr v, half hi -> row = v + 8*hi, col = lo
    float* Cp = C + (row0 + 8 * hi) * (long)D_ + n0 + lo;
#pragma unroll
    for (int v = 0; v < 8; ++v) {
        float val = c[v] + bv;
        if (RELU) val = fmaxf(val, 0.0f);
        Cp[(long)v * D_] = val;
    }
}

// ---------------- edge scatter: agg[dst] += hw[src] * dinv[src]*dinv[dst] ----------------
// one thread per (edge, feature); consecutive threads -> consecutive features (coalesced)
__global__ __launch_bounds__(256) void k_scatter(const int* __restrict__ src,
                                                 const int* __restrict__ dst,
                                                 const float* __restrict__ dinv,
                                                 const float* __restrict__ hw,
                                                 float* __restrict__ agg) {
    const long t = (long)blockIdx.x * blockDim.x + threadIdx.x;   // < E_*128 exactly
    const int  e = (int)(t >> 7);
    const int  f = (int)(t & 127);
    const int  s = src[e];
    const int  d = dst[e];
    const float w = dinv[s] * dinv[d];
    atomicAdd(&agg[(long)d * D_ + f], hw[(long)s * D_ + f] * w);
}

// self-loop term + bias (+relu): out = agg + hw*dinv^2 + b
template <bool RELU>
__global__ __launch_bounds__(256) void k_finalize(const float* __restrict__ agg,
                                                  const float* __restrict__ hw,
                                                  const float* __restrict__ dinv,
                                                  const float* __restrict__ b,
                                                  float* __restrict__ out) {
    const long t  = (long)blockIdx.x * blockDim.x + threadIdx.x;  // < NN_*128 exactly
    const int  i  = (int)(t >> 7);
    const int  f  = (int)(t & 127);
    const float di = dinv[i];
    float v = agg[t] + hw[t] * di * di + b[f];
    if (RELU) v = fmaxf(v, 0.0f);
    out[t] = v;
}

// ---------------- edge predictor MLP ----------------
// out[p] = relu(concat(h[s],h[d]) @ Wp1 + bp1) @ Wp2 + bp2
__global__ __launch_bounds__(256) void k_pred(const float* __restrict__ h,
                                              const int* __restrict__ srcn,
                                              const int* __restrict__ dstn,
                                              const float* __restrict__ Wp1,
                                              const float* __restrict__ bp1,
                                              const float* __restrict__ Wp2,
                                              const float* __restrict__ bp2,
                                              float* __restrict__ out, int p_total) {
    __shared__ float sW[2 * D_ * H_];  // 64 KB: Wp1 [256,64] row-major
    for (int i = threadIdx.x; i < 2 * D_ * H_; i += 256) sW[i] = Wp1[i];
    __syncthreads();

    const int p = blockIdx.x * 256 + threadIdx.x;
    if (p >= p_total) return;

    float acc[H_];
#pragma unroll
    for (int j = 0; j < H_; ++j) acc[j] = bp1[j];   // uniform loads, cache-resident

    const float4* r1 = (const float4*)(h + (long)srcn[p] * D_);
    const float4* r2 = (const float4*)(h + (long)dstn[p] * D_);

    for (int k4 = 0; k4 < D_ / 4; ++k4) {
        const float4 v = r1[k4];
        const float* w = &sW[(k4 * 4) * H_];
#pragma unroll
        for (int j = 0; j < H_; ++j)
            acc[j] += v.x * w[j] + v.y * w[j + H_] + v.z * w[j + 2 * H_] + v.w * w[j + 3 * H_];
    }
    for (int k4 = 0; k4 < D_ / 4; ++k4) {
        const float4 v = r2[k4];
        const float* w = &sW[(D_ + k4 * 4) * H_];
#pragma unroll
        for (int j = 0; j < H_; ++j)
            acc[j] += v.x * w[j] + v.y * w[j + H_] + v.z * w[j + 2 * H_] + v.w * w[j + 3 * H_];
    }

    float o = bp2[0];
#pragma unroll
    for (int j = 0; j < H_; ++j) o += fmaxf(acc[j], 0.0f) * Wp2[j];
    out[p] = o;
}

// ---------------- host-side launch ----------------
extern "C" void kernel_launch(void* const* d_in, const int* in_sizes, int n_in,
                              void* d_out, int out_size, void* d_ws, size_t ws_size,
                              hipStream_t stream) {
    const int*   ei       = (const int*)d_in[0];     // [2,E]
    const float* x        = (const float*)d_in[1];   // [N,512]
    const int*   src_n    = (const int*)d_in[2];     // [P]
    const int*   dst_n    = (const int*)d_in[3];     // [P]
    const float* user_emb = (const float*)d_in[4];   // [NU,128]
    const float* Wf       = (const float*)d_in[5];   // [512,128]
    const float* bf       = (const float*)d_in[6];
    const float* W1       = (const float*)d_in[7];   // [128,128]
    const float* b1       = (const float*)d_in[8];
    const float* W2       = (const float*)d_in[9];
    const float* b2       = (const float*)d_in[10];
    const float* Wp1      = (const float*)d_in[11];  // [256,64]
    const float* bp1      = (const float*)d_in[12];
    const float* Wp2      = (const float*)d_in[13];  // [64,1]
    const float* bp2      = (const float*)d_in[14];
    float*       out      = (float*)d_out;

    const int* srcE = ei;        // edge_index[0]
    const int* dstE = ei + E_;   // edge_index[1]

    // workspace layout (floats): dinv | hA (h0/h1/h2) | hw | agg   ~ 462 MB
    float* ws   = (float*)d_ws;
    float* dinv = ws;                              // NN_
    float* hA   = ws + 300032;                     // NN_*128
    float* hw   = hA + (long)NN_ * D_;             // NN_*128
    float* agg  = hw + (long)NN_ * D_;             // NN_*128

    const long nh = (long)NN_ * D_;                 // 38.4M, divisible by 256

    // --- degree / normalization ---
    k_zero<<<(NN_ + 255) / 256, 256, 0, stream>>>(dinv, NN_);
    k_deg<<<(E_ + 255) / 256, 256, 0, stream>>>(dstE, dinv, E_);
    k_dinv<<<(NN_ + 255) / 256, 256, 0, stream>>>(dinv, NN_);

    // --- h0 = concat(user_emb, x_books @ Wf + bf) ---
    hipMemcpyAsync(hA, user_emb, (size_t)NU_ * D_ * sizeof(float),
                   hipMemcpyDeviceToDevice, stream);
    gemm16_wmma<F_, false, true><<<NB_ / 16, 256, 0, stream>>>(
        x + (long)NU_ * F_, Wf, bf, hA + (long)NU_ * D_);

    // --- GCN layer 1: h1 = relu(agg(h0 @ W1) + b1) ---
    gemm16_wmma<D_, false, false><<<NN_ / 16, 256, 0, stream>>>(hA, W1, nullptr, hw);
    k_zero<<<(int)(nh / 256), 256, 0, stream>>>(agg, nh);
    k_scatter<<<(E_ * D_) / 256, 256, 0, stream>>>(srcE, dstE, dinv, hw, agg);
    k_finalize<true><<<(int)(nh / 256), 256, 0, stream>>>(agg, hw, dinv, b1, hA);

    // --- GCN layer 2: h2 = agg(h1 @ W2) + b2 ---
    gemm16_wmma<D_, false, false><<<NN_ / 16, 256, 0, stream>>>(hA, W2, nullptr, hw);
    k_zero<<<(int)(nh / 256), 256, 0, stream>>>(agg, nh);
    k_scatter<<<(E_ * D_) / 256, 256, 0, stream>>>(srcE, dstE, dinv, hw, agg);
    k_finalize<false><<<(int)(nh / 256), 256, 0, stream>>>(agg, hw, dinv, b2, hA);

    // --- edge predictor ---
    k_pred<<<(P_ + 255) / 256, 256, 0, stream>>>(hA, src_n, dst_n, Wp1, bp1, Wp2, bp2,
                                                 out, P_);
}